// DMF_5171140624760
// MI455X (gfx1250) — compile-verified
//
#include <hip/hip_runtime.h>

typedef float v2f __attribute__((ext_vector_type(2)));
typedef float v8f __attribute__((ext_vector_type(8)));

constexpr int NB_  = 8192;   // batch
constexpr int DIN_ = 10000;  // interaction vector length
constexpr int HID_ = 1024;   // hidden
constexpr int E_   = 128;    // embedding

constexpr int BM  = 128;     // M tile
constexpr int BN  = 128;     // N tile
constexpr int BK  = 32;      // K tile
constexpr int BKP = 36;      // padded LDS K stride: 144B rows -> 16B aligned for B128
                             // async writes; m*36 mod 64 covers 16 distinct banks.

__device__ __forceinline__ unsigned short f32_to_bf16(float f) {
    unsigned u = __float_as_uint(f);
    u += 0x7FFFu + ((u >> 16) & 1u);   // round-to-nearest-even
    return (unsigned short)(u >> 16);
}

// CDNA5 async global->LDS copy, 16B per lane, tracked by ASYNCcnt.
__device__ __forceinline__ void async_cp_b128(unsigned ldsOff, const float* g) {
    asm volatile("global_load_async_to_lds_b128 %0, %1, off"
                 :: "v"(ldsOff), "v"(g) : "memory");
}
__device__ __forceinline__ void wait_async0() {
    asm volatile("s_wait_asynccnt 0x0" ::: "memory");
}
__device__ __forceinline__ unsigned lds_off(const void* p) {
    return (unsigned)(size_t)p;   // generic LDS pointer: addr[31:0] is the LDS offset
}

// 8 K-steps of V_WMMA_F32_16X16X4_F32 over one staged BK=32 tile.
__device__ __forceinline__ void mma_tile(const float* A, const float* B,
                                         int waveM, int waveN, int nl, int kh,
                                         v8f acc[2][4]) {
#pragma unroll
    for (int kk = 0; kk < 8; ++kk) {
        const int kf = (kk << 2) + kh;
        v2f a0 = *(const v2f*)(A + (waveM * 32 +      nl) * BKP + kf);
        v2f a1 = *(const v2f*)(A + (waveM * 32 + 16 + nl) * BKP + kf);
#pragma unroll
        for (int j = 0; j < 4; ++j) {
            v2f b = *(const v2f*)(B + (waveN * 64 + j * 16 + nl) * BKP + kf);
            acc[0][j] = __builtin_amdgcn_wmma_f32_16x16x4_f32(
                false, a0, false, b, (short)0, acc[0][j], false, false);
            acc[1][j] = __builtin_amdgcn_wmma_f32_16x16x4_f32(
                false, a1, false, b, (short)0, acc[1][j], false, false);
        }
    }
}

// -------------------------------------------------------------------------
// Layer 1: H1 = relu(X @ W1 + b1), X:[NB,DIN] f32, W1:[DIN,HID] f32 -> H1 bf16
// Double-buffered LDS pipeline: A tile via async global->LDS B128 copies,
// W tile via register loads + transposed LDS stores.
// -------------------------------------------------------------------------
__global__ __launch_bounds__(256)
void dmf_layer1(const float* __restrict__ X,
                const float* __restrict__ W1,
                const float* __restrict__ B1,
                unsigned short* __restrict__ H1)
{
    __shared__ float sA[2][BM * BKP];
    __shared__ float sB[2][BN * BKP];

    const int tid   = threadIdx.x;
    const int lane  = tid & 31;
    const int wave  = tid >> 5;
    const int waveM = wave >> 1;          // 0..3
    const int waveN = wave & 1;           // 0..1
    const int nl    = lane & 15;
    const int kh    = (lane >> 4) << 1;   // 0 or 2
    const int rAdd  = (lane >> 4) << 3;   // 0 or 8

    const int mBase = blockIdx.y * BM;
    const int nBase = blockIdx.x * BN;

    // per-thread staging coordinates
    const int aR  = tid >> 1;             // A: row 0..127 (2 threads/row)
    const int aC  = (tid & 1) << 4;       // A: k 0 or 16  (2 x B128 per thread)
    const int bKr = tid >> 5;             // B: k row  (0..7, 8 rows per pass, x4 passes)
    // B staging: q = tid + i*256, kr = q>>5 (0..31), c4 = (q&31)*4 (0..124)

    v8f acc[2][4];
#pragma unroll
    for (int i = 0; i < 2; ++i)
#pragma unroll
        for (int j = 0; j < 4; ++j)
            acc[i][j] = (v8f)0.0f;

    const float* aSrc0 = X + (size_t)(mBase + aR) * DIN_ + aC;       // + k0 per chunk
    const float* aSrc1 = aSrc0 + 4;                                  // second B128
    const unsigned aDst0 = lds_off(&sA[0][aR * BKP + aC]);
    const unsigned aDst1 = aDst0 + 16;
    const unsigned bufStride = (unsigned)(BM * BKP * sizeof(float)); // sA[1]-sA[0]

    constexpr int FULL = DIN_ / BK;          // 312 full chunks
    constexpr int TAILK = DIN_ - FULL * BK;  // 16

    float4 fB[4];
    // ---- prologue: stage chunk 0 into buffer 0 ----
    async_cp_b128(aDst0, aSrc0);
    async_cp_b128(aDst1, aSrc1);
#pragma unroll
    for (int i = 0; i < 4; ++i) {
        const int q  = tid + i * 256;
        const int kr = q >> 5;
        const int c4 = (q & 31) << 2;
        fB[i] = *(const float4*)(W1 + (size_t)kr * HID_ + nBase + c4);
    }
#pragma unroll
    for (int i = 0; i < 4; ++i) {
        const int q  = tid + i * 256;
        const int kr = q >> 5;
        const int c4 = (q & 31) << 2;
        float* d = &sB[0][0];
        d[(c4 + 0) * BKP + kr] = fB[i].x;
        d[(c4 + 1) * BKP + kr] = fB[i].y;
        d[(c4 + 2) * BKP + kr] = fB[i].z;
        d[(c4 + 3) * BKP + kr] = fB[i].w;
    }
    wait_async0();
    __syncthreads();

    // ---- main pipeline over 312 full chunks ----
    for (int kc = 0; kc < FULL; ++kc) {
        const int cur = kc & 1;
        const int nxt = cur ^ 1;
        const bool more = (kc + 1) < FULL;
        if (more) {
            const int k1 = (kc + 1) * BK;
            async_cp_b128(aDst0 + (unsigned)nxt * bufStride, aSrc0 + k1);
            async_cp_b128(aDst1 + (unsigned)nxt * bufStride, aSrc1 + k1);
#pragma unroll
            for (int i = 0; i < 4; ++i) {
                const int q  = tid + i * 256;
                const int kr = q >> 5;
                const int c4 = (q & 31) << 2;
                fB[i] = *(const float4*)(W1 + (size_t)(k1 + kr) * HID_ + nBase + c4);
            }
        }

        mma_tile(&sA[cur][0], &sB[cur][0], waveM, waveN, nl, kh, acc);

        if (more) {
#pragma unroll
            for (int i = 0; i < 4; ++i) {
                const int q  = tid + i * 256;
                const int kr = q >> 5;
                const int c4 = (q & 31) << 2;
                float* d = &sB[nxt][0];
                d[(c4 + 0) * BKP + kr] = fB[i].x;
                d[(c4 + 1) * BKP + kr] = fB[i].y;
                d[(c4 + 2) * BKP + kr] = fB[i].z;
                d[(c4 + 3) * BKP + kr] = fB[i].w;
            }
            wait_async0();
        }
        __syncthreads();
    }

    // ---- peeled tail chunk (16 valid K), zero-padded, register path ----
    {
        const int k0 = FULL * BK;   // 9984
#pragma unroll
        for (int i = 0; i < 4; ++i) {
            const int q  = tid + i * 256;
            const int r  = q >> 3;
            const int c4 = (q & 7) << 2;
            float4 f = make_float4(0.f, 0.f, 0.f, 0.f);
            if (c4 + 3 < TAILK)
                f = *(const float4*)(X + (size_t)(mBase + r) * DIN_ + k0 + c4);
            *(float4*)(&sA[0][r * BKP + c4]) = f;
        }
#pragma unroll
        for (int i = 0; i < 4; ++i) {
            const int q  = tid + i * 256;
            const int kr = q >> 5;
            const int c4 = (q & 31) << 2;
            float4 f = make_float4(0.f, 0.f, 0.f, 0.f);
            if (kr < TAILK)
                f = *(const float4*)(W1 + (size_t)(k0 + kr) * HID_ + nBase + c4);
            float* d = &sB[0][0];
            d[(c4 + 0) * BKP + kr] = f.x;
            d[(c4 + 1) * BKP + kr] = f.y;
            d[(c4 + 2) * BKP + kr] = f.z;
            d[(c4 + 3) * BKP + kr] = f.w;
        }
        __syncthreads();
        mma_tile(&sA[0][0], &sB[0][0], waveM, waveN, nl, kh, acc);
    }

    // ---- epilogue: +bias, ReLU, bf16 store ----
#pragma unroll
    for (int j = 0; j < 4; ++j) {
        const int col   = nBase + waveN * 64 + j * 16 + nl;
        const float bia = B1[col];
#pragma unroll
        for (int i = 0; i < 2; ++i) {
            const int mrow = mBase + waveM * 32 + i * 16 + rAdd;
#pragma unroll
            for (int v = 0; v < 8; ++v) {
                float val = acc[i][j][v] + bia;
                val = fmaxf(val, 0.f);
                H1[(size_t)(mrow + v) * HID_ + col] = f32_to_bf16(val);
            }
        }
    }
}

// -------------------------------------------------------------------------
// Head: p = Hu @ Wu2 + bu2, q = Hi @ Wi2 + bi2, out[m] = dot(p[m], q[m]).
// -------------------------------------------------------------------------
__global__ __launch_bounds__(256)
void dmf_head(const unsigned short* __restrict__ Hu,
              const float* __restrict__ Wu2, const float* __restrict__ bu2,
              const unsigned short* __restrict__ Hi,
              const float* __restrict__ Wi2, const float* __restrict__ bi2,
              float* __restrict__ out)
{
    __shared__ float sA[BM * BKP];
    __shared__ float sB[E_ * BKP];
    __shared__ float sRow[BM];

    const int tid   = threadIdx.x;
    const int lane  = tid & 31;
    const int wave  = tid >> 5;
    const int waveM = wave >> 1;
    const int waveN = wave & 1;
    const int nl    = lane & 15;
    const int kh    = (lane >> 4) << 1;
    const int rAdd  = (lane >> 4) << 3;

    const int mBase = blockIdx.x * BM;

    if (tid < BM) sRow[tid] = 0.f;

    v8f acc[2][2][4];   // [side][i][j]
#pragma unroll
    for (int s = 0; s < 2; ++s)
#pragma unroll
        for (int i = 0; i < 2; ++i)
#pragma unroll
            for (int j = 0; j < 4; ++j)
                acc[s][i][j] = (v8f)0.0f;

#pragma unroll
    for (int side = 0; side < 2; ++side) {
        const unsigned short* Hs = side ? Hi : Hu;
        const float*          W2 = side ? Wi2 : Wu2;

        for (int kc = 0; kc < HID_ / BK; ++kc) {
            const int k0 = kc * BK;
            // stage A tile: bf16 -> f32
#pragma unroll
            for (int i = 0; i < 2; ++i) {
                const int q  = tid + i * 256;     // 0..511
                const int r  = q >> 2;            // 0..127
                const int c8 = (q & 3) << 3;      // 0,8,16,24
                const uint4 hv = *(const uint4*)(Hs + (size_t)(mBase + r) * HID_ + k0 + c8);
                float* dst = sA + r * BKP + c8;
                dst[0] = __uint_as_float(hv.x << 16);
                dst[1] = __uint_as_float(hv.x & 0xFFFF0000u);
                dst[2] = __uint_as_float(hv.y << 16);
                dst[3] = __uint_as_float(hv.y & 0xFFFF0000u);
                dst[4] = __uint_as_float(hv.z << 16);
                dst[5] = __uint_as_float(hv.z & 0xFFFF0000u);
                dst[6] = __uint_as_float(hv.w << 16);
                dst[7] = __uint_as_float(hv.w & 0xFFFF0000u);
            }
            // stage B tile transposed
#pragma unroll
            for (int i = 0; i < 4; ++i) {
                const int q  = tid + i * 256;
                const int kr = q >> 5;            // 0..31
                const int c4 = (q & 31) << 2;     // 0..124
                const float4 f = *(const float4*)(W2 + (size_t)(k0 + kr) * E_ + c4);
                sB[(c4 + 0) * BKP + kr] = f.x;
                sB[(c4 + 1) * BKP + kr] = f.y;
                sB[(c4 + 2) * BKP + kr] = f.z;
                sB[(c4 + 3) * BKP + kr] = f.w;
            }
            __syncthreads();
            mma_tile(sA, sB, waveM, waveN, nl, kh, acc[side]);
            __syncthreads();
        }
    }

    float bp[4], bq[4];
#pragma unroll
    for (int j = 0; j < 4; ++j) {
        const int col = waveN * 64 + j * 16 + nl;
        bp[j] = bu2[col];
        bq[j] = bi2[col];
    }

#pragma unroll
    for (int i = 0; i < 2; ++i) {
#pragma unroll
        for (int v = 0; v < 8; ++v) {
            const int mloc = waveM * 32 + i * 16 + rAdd + v;
            float part = 0.f;
#pragma unroll
            for (int j = 0; j < 4; ++j)
                part += (acc[0][i][j][v] + bp[j]) * (acc[1][i][j][v] + bq[j]);
            atomicAdd(&sRow[mloc], part);
        }
    }
    __syncthreads();
    if (tid < BM) out[mBase + tid] = sRow[tid];
}

// -------------------------------------------------------------------------
extern "C" void kernel_launch(void* const* d_in, const int* in_sizes, int n_in,
                              void* d_out, int out_size, void* d_ws, size_t ws_size,
                              hipStream_t stream) {
    (void)in_sizes; (void)n_in; (void)out_size; (void)ws_size;

    const float* user = (const float*)d_in[0];
    const float* item = (const float*)d_in[1];
    const float* Wu1  = (const float*)d_in[2];
    const float* bu1  = (const float*)d_in[3];
    const float* Wu2  = (const float*)d_in[4];
    const float* bu2  = (const float*)d_in[5];
    const float* Wi1  = (const float*)d_in[6];
    const float* bi1  = (const float*)d_in[7];
    const float* Wi2  = (const float*)d_in[8];
    const float* bi2  = (const float*)d_in[9];
    float* out = (float*)d_out;

    unsigned short* Hu = (unsigned short*)d_ws;                 // [NB, HID] bf16
    unsigned short* Hi = Hu + (size_t)NB_ * HID_;               // [NB, HID] bf16

    dim3 g1(HID_ / BN, NB_ / BM);   // (8, 64)
    dmf_layer1<<<g1, 256, 0, stream>>>(user, Wu1, bu1, Hu);
    dmf_layer1<<<g1, 256, 0, stream>>>(item, Wi1, bi1, Hi);
    dmf_head<<<NB_ / BM, 256, 0, stream>>>(Hu, Wu2, bu2, Hi, Wi2, bi2, out);
}